// AdaptivePool_43946105372804
// MI455X (gfx1250) — compile-verified
//
#include <hip/hip_runtime.h>

#define SPATIAL_SCALE 0.0625f
#define POUT 28
#define CCH  256
#define HF   50
#define WF   76
#define RROI 128

// Padded tile dims
#define AP 32   // pooled rows padded (M of GEMM1, M of GEMM2)
#define HP 64   // feature-H padded (K of GEMM1), zero-filled 50..63
#define WP 80   // feature-W padded (N of GEMM1 / K of GEMM2), zero-filled 76..79
#define QP 32   // pooled cols padded (N of GEMM2)

#define WAVES 8 // one channel per wave

typedef __attribute__((ext_vector_type(2))) float v2f;
typedef __attribute__((ext_vector_type(8))) float v8f;

__device__ __forceinline__ v8f wmma_f32_4(v2f a, v2f b, v8f c) {
    // D = A(16x4,f32) * B(4x16,f32) + C(16x16,f32)
    return __builtin_amdgcn_wmma_f32_16x16x4_f32(
        false, a, false, b, (short)0, c, false, false);
}

__global__ __launch_bounds__(256)
void roi_adaptive_pool_wmma(const float* __restrict__ input,
                            const float* __restrict__ rois,
                            float* __restrict__ out)
{
    __shared__ float sA [AP * HP];          // A weights [p][h], zero padded (8 KB)
    __shared__ float sBt[WP * QP];          // B^T        [w][q], zero padded (10 KB)
    __shared__ float sTmp[WAVES][AP * 16];  // per-wave tmp strip [p][16]    (16 KB)

    const int blk  = blockIdx.x;
    const int r    = blk / (CCH / WAVES);
    const int cblk = blk % (CCH / WAVES);
    const int tid  = threadIdx.x;
    const int lane = tid & 31;
    const int wave = tid >> 5;
    const int c    = cblk * WAVES + wave;

    // ---- ROI box -> integer pooling windows (replicates reference exactly) ----
    const float* roi = rois + r * 5;
    const int   nidx = (int)roi[0];
    const float x1 = roi[1], y1 = roi[2], x2 = roi[3], y2 = roi[4];

    int lo_y = (int)floorf(SPATIAL_SCALE * y1 - 0.5f); lo_y = lo_y > 0 ? lo_y : 0;
    int hi_y = (int)ceilf (SPATIAL_SCALE * y2 - 0.5f);
    if (hi_y == lo_y) hi_y += 1;
    hi_y = hi_y < HF ? hi_y : HF;
    const int len_y = hi_y - lo_y;

    int lo_x = (int)floorf(SPATIAL_SCALE * x1 - 0.5f); lo_x = lo_x > 0 ? lo_x : 0;
    int hi_x = (int)ceilf (SPATIAL_SCALE * x2 - 0.5f);
    if (hi_x == lo_x) hi_x += 1;
    hi_x = hi_x < WF ? hi_x : WF;
    const int len_x = hi_x - lo_x;

    // ---- build pooling weight matrices in LDS ----
    for (int i = tid; i < AP * HP; i += 256) {
        const int p = i / HP, h = i % HP;
        float v = 0.0f;
        if (p < POUT && h < HF) {
            const int s = lo_y + (p * len_y) / POUT;
            const int e = lo_y + ((p + 1) * len_y + POUT - 1) / POUT;
            if (h >= s && h < e) v = 1.0f / (float)(e - s);
        }
        sA[i] = v;
    }
    for (int i = tid; i < WP * QP; i += 256) {
        const int w = i / QP, q = i % QP;
        float v = 0.0f;
        if (q < POUT && w < WF) {
            const int s = lo_x + (q * len_x) / POUT;
            const int e = lo_x + ((q + 1) * len_x + POUT - 1) / POUT;
            if (w >= s && w < e) v = 1.0f / (float)(e - s);
        }
        sBt[i] = v;
    }
    __syncthreads();

    const float* feat = input + ((size_t)nidx * CCH + c) * (HF * WF);
    float* sT = &sTmp[wave][0];

    const int hlf = lane >> 4;   // half-wave select (K split)
    const int l16 = lane & 15;

    v8f acc[2][2] = {};          // 28x28 output, 2x2 tiles of 16x16

    for (int w0 = 0; w0 < WP; w0 += 16) {
        // ---- GEMM1 strip: tmp[0:32][w0:w0+16] = A(32x52) * feat(52 x 16slice) ----
        v8f t0 = {}, t1 = {};
        for (int k0 = 0; k0 < 52; k0 += 4) {
            const int krow = k0 + 2 * hlf;
            const int wcol = w0 + l16;
            v2f bf;
            bf.x = (krow     < HF && wcol < WF) ? feat[ krow      * WF + wcol] : 0.0f;
            bf.y = (krow + 1 < HF && wcol < WF) ? feat[(krow + 1) * WF + wcol] : 0.0f;
            const v2f a0 = *(const v2f*)&sA[ l16       * HP + krow];
            const v2f a1 = *(const v2f*)&sA[(16 + l16) * HP + krow];
            t0 = wmma_f32_4(a0, bf, t0);
            t1 = wmma_f32_4(a1, bf, t1);
        }
        // ---- re-layout strip through per-wave LDS (C-layout -> A-operand) ----
        #pragma unroll
        for (int v = 0; v < 8; v++) {
            sT[( v + 8 * hlf     ) * 16 + l16] = t0[v];
            sT[(16 + v + 8 * hlf ) * 16 + l16] = t1[v];
        }
        // same-wave LDS ops are in-order (DScnt) -> no barrier needed
        // ---- GEMM2 accumulate: acc += tmp_strip(32x16) * Bt[w0:w0+16][0:32] ----
        for (int k0 = 0; k0 < 16; k0 += 4) {
            const int krow = k0 + 2 * hlf;
            const v2f ta0 = *(const v2f*)&sT[ l16       * 16 + krow];
            const v2f ta1 = *(const v2f*)&sT[(16 + l16) * 16 + krow];
            const int wr = w0 + krow;
            v2f b0, b1;
            b0.x = sBt[ wr      * QP +      l16];
            b0.y = sBt[(wr + 1) * QP +      l16];
            b1.x = sBt[ wr      * QP + 16 + l16];
            b1.y = sBt[(wr + 1) * QP + 16 + l16];
            acc[0][0] = wmma_f32_4(ta0, b0, acc[0][0]);
            acc[0][1] = wmma_f32_4(ta0, b1, acc[0][1]);
            acc[1][0] = wmma_f32_4(ta1, b0, acc[1][0]);
            acc[1][1] = wmma_f32_4(ta1, b1, acc[1][1]);
        }
    }

    // ---- write 28x28 block: out[r][c][p][q] ----
    const size_t base = ((size_t)r * CCH + c) * (POUT * POUT);
    #pragma unroll
    for (int mt = 0; mt < 2; mt++) {
        #pragma unroll
        for (int qt = 0; qt < 2; qt++) {
            #pragma unroll
            for (int v = 0; v < 8; v++) {
                const int p = mt * 16 + v + 8 * hlf;
                const int q = qt * 16 + l16;
                if (p < POUT && q < POUT)
                    out[base + p * POUT + q] = acc[mt][qt][v];
            }
        }
    }
}

extern "C" void kernel_launch(void* const* d_in, const int* in_sizes, int n_in,
                              void* d_out, int out_size, void* d_ws, size_t ws_size,
                              hipStream_t stream) {
    const float* input = (const float*)d_in[0];   // [4, 256, 50, 76] fp32
    const float* rois  = (const float*)d_in[1];   // [128, 5] fp32
    float* out = (float*)d_out;                   // [128, 256, 28, 28] fp32
    (void)in_sizes; (void)n_in; (void)out_size; (void)d_ws; (void)ws_size;

    const int blocks = RROI * (CCH / WAVES);      // 4096 blocks, 256 thr (8 waves)
    roi_adaptive_pool_wmma<<<blocks, 256, 0, stream>>>(input, rois, out);
}